// HGCNEncoder_41644002902694
// MI455X (gfx1250) — compile-verified
//
#include <hip/hip_runtime.h>
#include <hip/hip_bf16.h>

typedef __attribute__((ext_vector_type(2))) float v2f;
typedef __attribute__((ext_vector_type(8))) float v8f;

// ---------------------------------------------------------------------------
// Zero-fill helper (graph-capture safe alternative to memset)
// ---------------------------------------------------------------------------
__global__ void hgcn_zero_f32(float* __restrict__ p, int n) {
    int i = blockIdx.x * blockDim.x + threadIdx.x;
    if (i < n) p[i] = 0.0f;
}

// ---------------------------------------------------------------------------
// Degree accumulation: deg[node]++ , bdeg[edge]++
// ---------------------------------------------------------------------------
__global__ void hgcn_degree(const int* __restrict__ ni, const int* __restrict__ ei,
                            float* __restrict__ deg, float* __restrict__ bdeg, int P) {
    int p = blockIdx.x * blockDim.x + threadIdx.x;
    if (p < P) {
        atomicAdd(&deg[ni[p]], 1.0f);
        atomicAdd(&bdeg[ei[p]], 1.0f);
    }
}

__global__ void hgcn_invert(float* __restrict__ v, int n) {
    int i = blockIdx.x * blockDim.x + threadIdx.x;
    if (i < n) { float d = v[i]; v[i] = d > 0.0f ? 1.0f / d : 0.0f; }
}

// ---------------------------------------------------------------------------
// FP32 WMMA GEMM: Y[nrows,64] = X[nrows,64] @ W[64,64]
// One wave computes one 16x16 tile via V_WMMA_F32_16X16X4_F32 (16 chained
// accumulating WMMAs over K=64). Block = 256 threads = 8 waves = 32 rows.
// X tile (8 KB) and W (16 KB) staged in LDS with b128 loads; out-of-range
// rows are CLAMPED (not zeroed): garbage only reaches output rows whose
// store is guarded, since out[m,*] depends solely on A row m.
//
// ISA layouts (CDNA5 ISA §7.12.2):
//   A 16x4 f32 : lanes 0-15 rows M=0..15 hold {K=0,K=1}; lanes 16-31 {K=2,K=3}
//   B 4x16 f32 : VGPR0 lanes0-15 K=0 / lanes16-31 K=2; VGPR1 K=1 / K=3
//   C/D 16x16  : VGPR r -> lanes0-15 (M=r), lanes16-31 (M=r+8), N=lane%16
// ---------------------------------------------------------------------------
__global__ __launch_bounds__(256) void hgcn_gemm64_wmma(
    const float* __restrict__ X, const float* __restrict__ W,
    float* __restrict__ Y, int nrows) {
    __shared__ float Ws[64 * 64];   // 16 KB
    __shared__ float Xs[32 * 64];   //  8 KB

    // stage W: 4096 floats = 1024 float4
    {
        const float4* __restrict__ W4 = (const float4*)W;
        float4* Ws4 = (float4*)Ws;
        for (int i = threadIdx.x; i < 1024; i += 256) Ws4[i] = W4[i];
    }
    // stage X tile: 32 rows x 64 floats = 512 float4 (rows clamped in-bounds)
    {
        const int rowStart = blockIdx.x * 32;
        float4* Xs4 = (float4*)Xs;
        for (int i = threadIdx.x; i < 512; i += 256) {
            const int r  = i >> 4;    // local row 0..31
            const int cc = i & 15;    // float4 column 0..15
            const int gr = min(rowStart + r, nrows - 1);
            Xs4[i] = ((const float4*)(X + (size_t)gr * 64))[cc];
        }
    }
    __syncthreads();

    const int wave = threadIdx.x >> 5;
    const int lane = threadIdx.x & 31;
    const int half = lane >> 4;          // 0: K pair {0,1}, 1: K pair {2,3}
    const int l16  = lane & 15;

    const int rowLoc  = (wave >> 2) * 16;      // 0 or 16 within tile
    const int colBase = (wave & 3) * 16;
    const int rowBase = blockIdx.x * 32 + rowLoc;

    const float* __restrict__ arow = Xs + (size_t)(rowLoc + l16) * 64;

    // preload all fragments, then issue 16 back-to-back accumulating WMMAs
    v2f afrag[16], bfrag[16];
#pragma unroll
    for (int j = 0; j < 16; ++j) {
        const int ka = 4 * j + half * 2;
        afrag[j].x = arow[ka];
        afrag[j].y = arow[ka + 1];
        bfrag[j].x = Ws[(ka)     * 64 + colBase + l16];
        bfrag[j].y = Ws[(ka + 1) * 64 + colBase + l16];
    }
    v8f c = {};
#pragma unroll
    for (int j = 0; j < 16; ++j) {
        c = __builtin_amdgcn_wmma_f32_16x16x4_f32(
                /*neg_a=*/false, afrag[j], /*neg_b=*/false, bfrag[j],
                /*c_mod=*/(short)0, c, /*reuse_a=*/false, /*reuse_b=*/false);
    }

#pragma unroll
    for (int r = 0; r < 8; ++r) {
        const int m = rowBase + r + half * 8;
        if (m < nrows) Y[(size_t)m * 64 + colBase + l16] = c[r];
    }
}

// ---------------------------------------------------------------------------
// Scatter-add of 64-float rows: dst[didx[p]] += src[sidx[p]] * (sscale?sscale[sidx[p]]:1)
// 16 lanes per pair, one float4 (global_load_b128) per lane -> coalesced 256B row read.
// Destinations stay L2-resident (192 MB L2 >> working set) -> L2 atomic bound.
// ---------------------------------------------------------------------------
__global__ __launch_bounds__(256) void hgcn_scatter_add(
    const float* __restrict__ src, const int* __restrict__ sidx,
    const int* __restrict__ didx, const float* __restrict__ sscale,
    float* __restrict__ dst, int P) {
    const int t   = blockIdx.x * 256 + threadIdx.x;
    const int p   = t >> 4;
    const int sub = t & 15;
    if (p >= P) return;
    const int s  = sidx[p];
    const int dI = didx[p];
    const float4 v = ((const float4*)(src + (size_t)s * 64))[sub];
    const float sc = sscale ? sscale[s] : 1.0f;
    float* dp = dst + (size_t)dI * 64 + sub * 4;
    atomicAdd(dp + 0, v.x * sc);
    atomicAdd(dp + 1, v.y * sc);
    atomicAdd(dp + 2, v.z * sc);
    atomicAdd(dp + 3, v.w * sc);
}

// ---------------------------------------------------------------------------
// Fused epilogue: h = [leaky_relu](h * Dinv[row] + bias[col])
// ---------------------------------------------------------------------------
__global__ void hgcn_epilogue(float* __restrict__ h, const float* __restrict__ dinv,
                              const float* __restrict__ bias, int n, int relu) {
    int i = blockIdx.x * blockDim.x + threadIdx.x;
    if (i >= n) return;
    const int row = i >> 6;
    const int col = i & 63;
    float v = h[i] * dinv[row] + bias[col];
    if (relu) v = v > 0.0f ? v : 0.01f * v;
    h[i] = v;
}

// ---------------------------------------------------------------------------
extern "C" void kernel_launch(void* const* d_in, const int* in_sizes, int n_in,
                              void* d_out, int out_size, void* d_ws, size_t ws_size,
                              hipStream_t stream) {
    const float* x        = (const float*)d_in[0];
    const int*   node_idx = (const int*)  d_in[1];
    const int*   edge_idx = (const int*)  d_in[2];
    const float* W0 = (const float*)d_in[3];
    const float* b0 = (const float*)d_in[4];
    const float* W1 = (const float*)d_in[5];
    const float* b1 = (const float*)d_in[6];
    const float* W2 = (const float*)d_in[7];
    const float* b2 = (const float*)d_in[8];
    float* out = (float*)d_out;

    const int N = in_sizes[0] / 64;   // 50000
    const int P = in_sizes[1];        // 800000
    const int E = out_size / 64;      // 10000

    // workspace layout (floats): deg[N] | bdeg[E] | xw[N*64] | e[E*64] | h[N*64]
    float* deg  = (float*)d_ws;
    float* bdeg = deg + N;
    float* xw   = bdeg + E;
    float* e    = xw + (size_t)N * 64;
    float* h    = e + (size_t)E * 64;

    const int NT = 256;
    const int gDegZ = (N + E + NT - 1) / NT;
    const int gP    = (P + NT - 1) / NT;
    const int gE64  = (E * 64 + NT - 1) / NT;
    const int gN64  = (N * 64 + NT - 1) / NT;
    const int gScat = (P * 16 + NT - 1) / NT;
    const int gGemm = (N + 31) / 32;

    // degrees (deg|bdeg contiguous -> single zero pass); inverted in place
    hgcn_zero_f32<<<gDegZ, NT, 0, stream>>>(deg, N + E);
    hgcn_degree<<<gP, NT, 0, stream>>>(node_idx, edge_idx, deg, bdeg, P);
    hgcn_invert<<<(N + NT - 1) / NT, NT, 0, stream>>>(deg, N);   // deg  -> D^-1
    hgcn_invert<<<(E + NT - 1) / NT, NT, 0, stream>>>(bdeg, E);  // bdeg -> B^-1

    // ---- layer 0: x @ W0 -> xw, hconv, leaky_relu -> h
    hgcn_gemm64_wmma<<<gGemm, NT, 0, stream>>>(x, W0, xw, N);
    hgcn_zero_f32<<<gE64, NT, 0, stream>>>(e, E * 64);
    hgcn_scatter_add<<<gScat, NT, 0, stream>>>(xw, node_idx, edge_idx, nullptr, e, P);
    hgcn_zero_f32<<<gN64, NT, 0, stream>>>(h, N * 64);
    hgcn_scatter_add<<<gScat, NT, 0, stream>>>(e, edge_idx, node_idx, bdeg, h, P);
    hgcn_epilogue<<<gN64, NT, 0, stream>>>(h, deg, b0, N * 64, 1);

    // ---- layer 1: h @ W1 -> xw, hconv, leaky_relu -> h
    hgcn_gemm64_wmma<<<gGemm, NT, 0, stream>>>(h, W1, xw, N);
    hgcn_zero_f32<<<gE64, NT, 0, stream>>>(e, E * 64);
    hgcn_scatter_add<<<gScat, NT, 0, stream>>>(xw, node_idx, edge_idx, nullptr, e, P);
    hgcn_zero_f32<<<gN64, NT, 0, stream>>>(h, N * 64);
    hgcn_scatter_add<<<gScat, NT, 0, stream>>>(e, edge_idx, node_idx, bdeg, h, P);
    hgcn_epilogue<<<gN64, NT, 0, stream>>>(h, deg, b1, N * 64, 1);

    // ---- layer 2: h @ W2 -> xw, hconv (no relu) -> h
    hgcn_gemm64_wmma<<<gGemm, NT, 0, stream>>>(h, W2, xw, N);
    hgcn_zero_f32<<<gE64, NT, 0, stream>>>(e, E * 64);
    hgcn_scatter_add<<<gScat, NT, 0, stream>>>(xw, node_idx, edge_idx, nullptr, e, P);
    hgcn_zero_f32<<<gN64, NT, 0, stream>>>(h, N * 64);
    hgcn_scatter_add<<<gScat, NT, 0, stream>>>(e, edge_idx, node_idx, bdeg, h, P);
    hgcn_epilogue<<<gN64, NT, 0, stream>>>(h, deg, b2, N * 64, 0);

    // ---- final pooling: out[E,64] = segment_sum(h[node_idx]) by edge_idx
    hgcn_zero_f32<<<gE64, NT, 0, stream>>>(out, E * 64);
    hgcn_scatter_add<<<gScat, NT, 0, stream>>>(h, node_idx, edge_idx, nullptr, out, P);
}